// Model_88648124989927
// MI455X (gfx1250) — compile-verified
//
#include <hip/hip_runtime.h>
#include <stdint.h>

// ============================================================================
// MI455X (gfx1250) implementation of the dual-branch GCN/DRSNet/GATv2 model.
//
// Roofline: ~90 GFLOP total vs ~2 GB of HBM traffic -> bandwidth bound at
// 23.3 TB/s (~80us floor). All dense GEMMs use the exact-precision f32 matrix
// op V_WMMA_F32_16X16X4_F32 (wave32, 16x16 tile per wave). GEMM operand tiles
// are staged with CDNA5 GLOBAL_LOAD_ASYNC_TO_LDS_B32 (ASYNCcnt-tracked, no
// VGPR round-trip), and LDS tiles are K-pair interleaved so each WMMA operand
// fragment is a single aligned ds_load_b64 into a consecutive VGPR pair.
// Edge scatter phases use f32 global atomics.
// ============================================================================

#define NN 8192
#define EE 262144
#define H1C 256
#define H2C 512
#define HEADS 4
#define OUTC 128

typedef float v2f __attribute__((ext_vector_type(2)));
typedef float v8f __attribute__((ext_vector_type(8)));

// --- CDNA5 async global->LDS copy (ASYNCcnt) -------------------------------
// LDS operand: low 32 bits of the flat shared-space pointer are the
// workgroup-relative LDS byte offset (ISA 10.2 aperture mapping).
__device__ __forceinline__ void async_stage_b32(const float* g, float* l) {
  asm volatile("global_load_async_to_lds_b32 %0, %1, off"
               :
               : "v"((unsigned)(uintptr_t)l), "v"(g)
               : "memory");
}
__device__ __forceinline__ void wait_async_all() {
  asm volatile("s_wait_asynccnt 0x0" ::: "memory");
}

// ---------------------------------------------------------------------------
// Generic f32 WMMA GEMM:  C[M,Nc] = A @ B (+ bias[n])
//   A(m,k) at A[m*lda_r + k*lda_c]   (supports transposed adjacency access)
//   B(k,n) at B[k*ldb_k + n*ldb_n]   (supports B^T for the final product)
//   C row-major, ld = Nc.
// Block: 128 threads = 4 waves; block tile 64x64; each wave owns a 16-row
// stripe and 4 column tiles of 16 (4x v8f accumulators). K staged in LDS in
// chunks of 16, stored as K-pairs (v2f) so A/B fragments per ISA 7.12.2
// (lanes 0-15 hold K={2p,2p+1}, lanes 16-31 K={2p+2,2p+3}) are single b64
// LDS loads. All M,Nc used here are multiples of 64, K multiples of 16.
// ---------------------------------------------------------------------------
#define GBM 64
#define GBN 64
#define GBK 16

__global__ __launch_bounds__(128)
void k_gemm_f32_wmma(const float* __restrict__ A, const float* __restrict__ B,
                     float* __restrict__ C, const float* __restrict__ bias,
                     int M, int Nc, int K,
                     long long lda_r, long long lda_c,
                     long long ldb_k, long long ldb_n) {
  __shared__ v2f Aps[GBK / 2][GBM];  // pair p holds k = {2p, 2p+1}
  __shared__ v2f Bps[GBK / 2][GBN];

  const int tid  = threadIdx.x;
  const int wave = tid >> 5;
  const int lane = tid & 31;
  const int half = lane >> 4;
  const int l16  = lane & 15;
  const int bm0  = blockIdx.y * GBM;
  const int bn0  = blockIdx.x * GBN;

  v8f acc[4];
  for (int t = 0; t < 4; ++t)
    for (int r = 0; r < 8; ++r) acc[t][r] = 0.0f;

  for (int k0 = 0; k0 < K; k0 += GBK) {
    // --- stage A tile (64 rows x 16 k) via async global->LDS ---
    if (lda_c == 1) {  // K contiguous in memory: let K vary fastest
      for (int i = tid; i < GBM * GBK; i += 128) {
        int k = i & (GBK - 1), m = i >> 4;
        async_stage_b32(&A[(long long)(bm0 + m) * lda_r + (long long)(k0 + k)],
                        ((float*)&Aps[k >> 1][m]) + (k & 1));
      }
    } else {           // M contiguous: let M vary fastest
      for (int i = tid; i < GBM * GBK; i += 128) {
        int m = i & (GBM - 1), k = i >> 6;
        async_stage_b32(&A[(long long)(bm0 + m) * lda_r + (long long)(k0 + k) * lda_c],
                        ((float*)&Aps[k >> 1][m]) + (k & 1));
      }
    }
    // --- stage B tile (16 k x 64 cols) ---
    if (ldb_n == 1) {
      for (int i = tid; i < GBK * GBN; i += 128) {
        int n = i & (GBN - 1), k = i >> 6;
        async_stage_b32(&B[(long long)(k0 + k) * ldb_k + (long long)(bn0 + n)],
                        ((float*)&Bps[k >> 1][n]) + (k & 1));
      }
    } else {
      for (int i = tid; i < GBK * GBN; i += 128) {
        int k = i & (GBK - 1), n = i >> 4;
        async_stage_b32(&B[(long long)(k0 + k) * ldb_k + (long long)(bn0 + n) * ldb_n],
                        ((float*)&Bps[k >> 1][n]) + (k & 1));
      }
    }
    wait_async_all();   // own wave's ASYNCcnt -> 0
    __syncthreads();    // all waves' LDS writes visible

    // --- compute: 4 wmma K-steps of 4 over this chunk, 4 column tiles ---
#pragma unroll
    for (int kk = 0; kk < GBK; kk += 4) {
      // A fragment (16x4): lanes 0-15 K={kk,kk+1}, lanes 16-31 K={kk+2,kk+3}
      v2f a = Aps[(kk >> 1) + half][wave * 16 + l16];
#pragma unroll
      for (int t = 0; t < 4; ++t) {
        v2f b = Bps[(kk >> 1) + half][t * 16 + l16];
        acc[t] = __builtin_amdgcn_wmma_f32_16x16x4_f32(
            false, a, false, b, (short)0, acc[t], false, false);
      }
    }
    __syncthreads();
  }

  // --- store: D layout = 8 VGPRs, VGPR r -> rows r (lanes 0-15) / r+8 ---
  for (int t = 0; t < 4; ++t) {
    int col = bn0 + t * 16 + l16;
    float bv = bias ? bias[col] : 0.0f;
#pragma unroll
    for (int r = 0; r < 8; ++r) {
      int row = bm0 + wave * 16 + r + half * 8;
      C[(long long)row * Nc + col] = acc[t][r] + bv;
    }
  }
}

// ---------------------------------------------------------------------------
// Utility kernels
// ---------------------------------------------------------------------------
__global__ void k_fill_f32(float* p, float v, long long n) {
  long long i = blockIdx.x * (long long)blockDim.x + threadIdx.x;
  if (i < n) p[i] = v;
}
__global__ void k_fill_u32(unsigned* p, unsigned v, long long n) {
  long long i = blockIdx.x * (long long)blockDim.x + threadIdx.x;
  if (i < n) p[i] = v;
}
__global__ void k_add(const float* __restrict__ a, const float* __restrict__ b,
                      float* __restrict__ c, long long n) {
  long long i = blockIdx.x * (long long)blockDim.x + threadIdx.x;
  if (i < n) c[i] = a[i] + b[i];
}
__global__ void k_reduce_sum(const float* __restrict__ v, float* out, int n) {
  __shared__ float sh[256];
  float s = 0.f;
  for (long long i = blockIdx.x * (long long)blockDim.x + threadIdx.x; i < n;
       i += (long long)gridDim.x * blockDim.x)
    s += v[i];
  sh[threadIdx.x] = s;
  __syncthreads();
  for (int st = 128; st > 0; st >>= 1) {
    if (threadIdx.x < st) sh[threadIdx.x] += sh[threadIdx.x + st];
    __syncthreads();
  }
  if (threadIdx.x == 0) atomicAdd(out, sh[0]);
}

// ---------------------------------------------------------------------------
// GCN: deg / symmetric-norm / weighted aggregation with self loops
// ---------------------------------------------------------------------------
__global__ void k_deg_edges(const int* __restrict__ dst, const float* __restrict__ w,
                            float* deg, int Ee) {
  int e = blockIdx.x * blockDim.x + threadIdx.x;
  if (e < Ee) atomicAdd(&deg[dst[e]], w[e]);
}
__global__ void k_dinv(float* deg, int n) {
  int i = blockIdx.x * blockDim.x + threadIdx.x;
  if (i < n) deg[i] = (deg[i] > 0.f) ? rsqrtf(deg[i]) : 0.f;
}
// out[i,c] = dinv[i]^2 * h[i,c]   (self-loop term; also initializes out)
__global__ void k_gcn_selfinit(const float* __restrict__ h, const float* __restrict__ dinv,
                               float* __restrict__ out, int logC, long long total) {
  long long i = blockIdx.x * (long long)blockDim.x + threadIdx.x;
  if (i >= total) return;
  long long node = i >> logC;
  float dv = dinv[node];
  out[i] = dv * dv * h[i];
}
// out[d,c] += dinv[s]*w*dinv[d] * h[s,c]
__global__ void k_gcn_agg(const float* __restrict__ h, const int* __restrict__ src,
                          const int* __restrict__ dst, const float* __restrict__ w,
                          const float* __restrict__ dinv, float* __restrict__ out,
                          int logC, long long total) {
  long long tid = blockIdx.x * (long long)blockDim.x + threadIdx.x;
  if (tid >= total) return;
  int C = 1 << logC;
  int c = (int)(tid & (C - 1));
  long long e = tid >> logC;
  int s = src[e], d = dst[e];
  float coef = dinv[s] * w[e] * dinv[d];
  atomicAdd(&out[(long long)d * C + c], coef * h[(long long)s * C + c]);
}
// x = leaky_relu(agg + bias, 0.01)
__global__ void k_gcn_final(const float* __restrict__ agg, const float* __restrict__ b,
                            float* __restrict__ out, int logC, long long total) {
  long long i = blockIdx.x * (long long)blockDim.x + threadIdx.x;
  if (i >= total) return;
  int c = (int)(i & ((1 << logC) - 1));
  float v = agg[i] + b[c];
  out[i] = v > 0.f ? v : 0.01f * v;
}

// ---------------------------------------------------------------------------
// DRSNet: whole network per node in one block, staged through LDS.
// ---------------------------------------------------------------------------
struct DrsBlockP {
  const float *bn1_b, *bn1_g, *bn2_b, *bn2_g, *bnf_b, *bnf_g;
  const float *c1_b, *c1_w, *c2_b, *c2_w, *fc1_b, *fc1_w, *fc2_b, *fc2_w;
};
struct DrsAll {
  DrsBlockP blk[6];
  const float *cls_b, *cls_w, *conv1_b, *conv1_w;
};

__device__ __forceinline__ void rsbu_block(float* X, float* Hh, float* Tt,
                                           float* gap, float* av, float* aa,
                                           const DrsBlockP& p, int cin, int cout,
                                           int L, int stride) {
  const int tid = threadIdx.x, nt = blockDim.x;
  const int Lo = L / stride;
  const float rr = rsqrtf(1.0f + 1e-5f);  // eval BN: running mean 0, var 1
  // bn1 + relu -> Tt[cin, L]
  for (int i = tid; i < cin * L; i += nt) {
    int c = i / L;
    float v = X[i] * (p.bn1_g[c] * rr) + p.bn1_b[c];
    Tt[i] = v > 0.f ? v : 0.f;
  }
  __syncthreads();
  // conv1 (k=3, pad=1, stride) -> Hh[cout, Lo]
  for (int i = tid; i < cout * Lo; i += nt) {
    int co = i / Lo, lo = i % Lo;
    float acc = p.c1_b[co];
    int base = lo * stride - 1;
    for (int ci = 0; ci < cin; ++ci) {
      const float* w = p.c1_w + (co * cin + ci) * 3;
      const float* t = Tt + ci * L;
      for (int k = 0; k < 3; ++k) {
        int idx = base + k;
        if (idx >= 0 && idx < L) acc += w[k] * t[idx];
      }
    }
    Hh[i] = acc;
  }
  __syncthreads();
  // bn2 + relu -> Tt[cout, Lo]
  for (int i = tid; i < cout * Lo; i += nt) {
    int c = i / Lo;
    float v = Hh[i] * (p.bn2_g[c] * rr) + p.bn2_b[c];
    Tt[i] = v > 0.f ? v : 0.f;
  }
  __syncthreads();
  // conv2 (k=3, pad=1, stride 1) -> Hh[cout, Lo]
  for (int i = tid; i < cout * Lo; i += nt) {
    int co = i / Lo, lo = i % Lo;
    float acc = p.c2_b[co];
    for (int ci = 0; ci < cout; ++ci) {
      const float* w = p.c2_w + (co * cout + ci) * 3;
      const float* t = Tt + ci * Lo;
      for (int k = 0; k < 3; ++k) {
        int idx = lo - 1 + k;
        if (idx >= 0 && idx < Lo) acc += w[k] * t[idx];
      }
    }
    Hh[i] = acc;
  }
  __syncthreads();
  // GAP of |h|
  if (tid < cout) {
    float s = 0.f;
    for (int l = 0; l < Lo; ++l) s += fabsf(Hh[tid * Lo + l]);
    gap[tid] = s / (float)Lo;
  }
  __syncthreads();
  // fc1 + bnf + relu
  if (tid < cout) {
    float s = p.fc1_b[tid];
    for (int j = 0; j < cout; ++j) s += gap[j] * p.fc1_w[j * cout + tid];
    s = s * (p.bnf_g[tid] * rr) + p.bnf_b[tid];
    av[tid] = s > 0.f ? s : 0.f;
  }
  __syncthreads();
  // fc2 + sigmoid -> channel threshold thr = gap * a
  if (tid < cout) {
    float s = p.fc2_b[tid];
    for (int j = 0; j < cout; ++j) s += av[j] * p.fc2_w[j * cout + tid];
    aa[tid] = gap[tid] / (1.f + __expf(-s));
  }
  __syncthreads();
  // soft threshold + shortcut (x[:, ::stride], channel zero-pad) -> Tt
  for (int i = tid; i < cout * Lo; i += nt) {
    int c = i / Lo, lo = i % Lo;
    float h = Hh[i];
    float r = fabsf(h) - aa[c];
    r = r > 0.f ? r : 0.f;
    float v = (h >= 0.f) ? r : -r;
    float sc = (c < cin) ? X[c * L + lo * stride] : 0.f;
    Tt[i] = v + sc;
  }
  __syncthreads();
  for (int i = tid; i < cout * Lo; i += nt) X[i] = Tt[i];
  __syncthreads();
}

__global__ __launch_bounds__(128)
void k_drsnet(const float* __restrict__ x2, float* __restrict__ x3, DrsAll P) {
  __shared__ float X[1024], Hh[1024], Tt[1024];
  __shared__ float gap[16], av[16], aa[16];
  const int node = blockIdx.x, tid = threadIdx.x, nt = blockDim.x;
  const float* xin = x2 + (long long)node * H2C;
  for (int i = tid; i < H2C; i += nt) Hh[i] = xin[i];
  __syncthreads();
  // conv1: 1->4 channels, L 512->256, k=3 pad=1 stride=2
  for (int i = tid; i < 4 * 256; i += nt) {
    int co = i >> 8, lo = i & 255;
    float acc = P.conv1_b[co];
    const float* w = P.conv1_w + co * 3;
    int base = 2 * lo - 1;
    for (int k = 0; k < 3; ++k) {
      int idx = base + k;
      if (idx >= 0 && idx < 512) acc += w[k] * Hh[idx];
    }
    X[i] = acc;
  }
  __syncthreads();
  rsbu_block(X, Hh, Tt, gap, av, aa, P.blk[0], 4, 4, 256, 2);
  rsbu_block(X, Hh, Tt, gap, av, aa, P.blk[1], 4, 4, 128, 1);
  rsbu_block(X, Hh, Tt, gap, av, aa, P.blk[2], 4, 8, 128, 2);
  rsbu_block(X, Hh, Tt, gap, av, aa, P.blk[3], 8, 8, 64, 1);
  rsbu_block(X, Hh, Tt, gap, av, aa, P.blk[4], 8, 16, 64, 2);
  rsbu_block(X, Hh, Tt, gap, av, aa, P.blk[5], 16, 16, 32, 1);
  // GAP over L=32
  if (tid < 16) {
    float s = 0.f;
    for (int l = 0; l < 32; ++l) s += X[tid * 32 + l];
    gap[tid] = s * (1.f / 32.f);
  }
  __syncthreads();
  // Linear 16 -> 256, then leaky_relu(0.01) (applied in _branch)
  for (int o = tid; o < H1C; o += nt) {
    float s = P.cls_b[o];
    for (int c = 0; c < 16; ++c) s += gap[c] * P.cls_w[c * H1C + o];
    x3[(long long)node * H1C + o] = s > 0.f ? s : 0.01f * s;
  }
}

// ---------------------------------------------------------------------------
// GATv2: per-(edge,head) logits -> segment max -> exp/sum/accumulate -> mean
// ---------------------------------------------------------------------------
__device__ __forceinline__ unsigned enc_key(float f) {
  unsigned b = __float_as_uint(f);
  return (b & 0x80000000u) ? ~b : (b | 0x80000000u);
}
__device__ __forceinline__ float dec_key(unsigned k) {
  unsigned b = (k & 0x80000000u) ? (k & 0x7FFFFFFFu) : ~k;
  return __uint_as_float(b);
}

__global__ void k_gat_logit(const float* __restrict__ xl, const float* __restrict__ xr,
                            const int* __restrict__ src, const int* __restrict__ dst,
                            const float* __restrict__ ew, const float* __restrict__ ewsum,
                            const float* __restrict__ We, const float* __restrict__ att,
                            float* __restrict__ logit, unsigned* __restrict__ mmax) {
  long long t = blockIdx.x * (long long)blockDim.x + threadIdx.x;
  const long long total = (long long)(EE + NN) * HEADS;
  if (t >= total) return;
  int h = (int)(t & 3);
  long long e = t >> 2;
  int s, d;
  float ea;
  if (e < EE) { s = src[e]; d = dst[e]; ea = ew[e]; }
  else        { s = d = (int)(e - EE); ea = ewsum[0] * (1.0f / EE); }
  const float* pl = xl + (long long)s * (HEADS * OUTC) + h * OUTC;
  const float* pr = xr + (long long)d * (HEADS * OUTC) + h * OUTC;
  const float* pe = We + h * OUTC;
  const float* pa = att + h * OUTC;
  float acc = 0.f;
  for (int c = 0; c < OUTC; ++c) {
    float v = pl[c] + pr[c] + pe[c] * ea;
    v = v > 0.f ? v : 0.2f * v;  // GATv2 leaky 0.2
    acc += pa[c] * v;
  }
  logit[t] = acc;
  atomicMax(&mmax[d * HEADS + h], enc_key(acc));
}

__global__ void k_gat_accum(const float* __restrict__ xl, const int* __restrict__ src,
                            const int* __restrict__ dst, const float* __restrict__ logit,
                            const unsigned* __restrict__ mmax, float* __restrict__ ssum,
                            float* __restrict__ accg) {
  long long t = blockIdx.x * (long long)blockDim.x + threadIdx.x;
  const long long total = (long long)(EE + NN) * HEADS;
  if (t >= total) return;
  int h = (int)(t & 3);
  long long e = t >> 2;
  int s, d;
  if (e < EE) { s = src[e]; d = dst[e]; } else { s = d = (int)(e - EE); }
  float m = dec_key(mmax[d * HEADS + h]);
  float al = __expf(logit[t] - m);
  atomicAdd(&ssum[d * HEADS + h], al);
  const float* pl = xl + (long long)s * (HEADS * OUTC) + h * OUTC;
  float* pa = accg + ((long long)d * HEADS + h) * OUTC;
  for (int c = 0; c < OUTC; ++c) atomicAdd(&pa[c], al * pl[c]);
}

__global__ void k_gat_final(const float* __restrict__ accg, const float* __restrict__ ssum,
                            const float* __restrict__ gb, float* __restrict__ x4) {
  long long i = blockIdx.x * (long long)blockDim.x + threadIdx.x;
  if (i >= (long long)NN * OUTC) return;
  long long d = i >> 7;
  int c = (int)(i & (OUTC - 1));
  float v = 0.f;
  for (int h = 0; h < HEADS; ++h)
    v += accg[(d * HEADS + h) * OUTC + c] / ssum[d * HEADS + h];
  x4[i] = v * (1.0f / HEADS) + gb[c];
}

// ---------------------------------------------------------------------------
// Host side
// ---------------------------------------------------------------------------
// d_in layout: top level in reference-argument order; nested param dicts in
// JAX tree-flatten (alphabetical-key) order:
//   branch params (99 arrays): blocks[0..5] x {bn1_b,bn1_g,bn2_b,bn2_g,bnf_b,
//   bnf_g,c1_b,c1_w,c2_b,c2_w,fc1_b,fc1_w,fc2_b,fc2_w}, cls_b, cls_w,
//   conv1_b, conv1_w, gat_We, gat_Wl, gat_Wr, gat_att, gat_b, gat_bl, gat_br,
//   gcn1_W, gcn1_b, gcn2_W, gcn2_b
enum {
  IN_MAT = 0, IN_DIS_EI, IN_DRUG_EI, IN_DIS_EW, IN_DRUG_EW,
  P_DIS = 5, P_PER_BRANCH = 99, P_DRUG = P_DIS + P_PER_BRANCH,
  PB_CLS_B = 84, PB_CLS_W, PB_CONV1_B, PB_CONV1_W,
  PB_GAT_WE = 88, PB_GAT_WL, PB_GAT_WR, PB_GAT_ATT, PB_GAT_B, PB_GAT_BL, PB_GAT_BR,
  PB_GCN1_W = 95, PB_GCN1_B, PB_GCN2_W, PB_GCN2_B,
};

struct Ws {
  float *B0, *B1, *B2, *B3, *B4, *B5, *B6;
  float *dinv, *logit, *ssum, *ewsum, *x4dis, *x4drug;
  unsigned* mmax;
};

static inline void launch_gemm(const float* A, const float* B, float* C,
                               const float* bias, int M, int Nc, int K,
                               long long lda_r, long long lda_c,
                               long long ldb_k, long long ldb_n, hipStream_t s) {
  dim3 g(Nc / GBN, M / GBM);
  k_gemm_f32_wmma<<<g, 128, 0, s>>>(A, B, C, bias, M, Nc, K, lda_r, lda_c, ldb_k, ldb_n);
}

static inline int blks(long long n, int t) { return (int)((n + t - 1) / t); }

static DrsAll mk_drs(void* const* d_in, int base) {
  DrsAll A;
  auto F = [&](int idx) { return (const float*)d_in[base + idx]; };
  for (int b = 0; b < 6; ++b) {
    int o = b * 14;
    A.blk[b].bn1_b = F(o + 0);  A.blk[b].bn1_g = F(o + 1);
    A.blk[b].bn2_b = F(o + 2);  A.blk[b].bn2_g = F(o + 3);
    A.blk[b].bnf_b = F(o + 4);  A.blk[b].bnf_g = F(o + 5);
    A.blk[b].c1_b  = F(o + 6);  A.blk[b].c1_w  = F(o + 7);
    A.blk[b].c2_b  = F(o + 8);  A.blk[b].c2_w  = F(o + 9);
    A.blk[b].fc1_b = F(o + 10); A.blk[b].fc1_w = F(o + 11);
    A.blk[b].fc2_b = F(o + 12); A.blk[b].fc2_w = F(o + 13);
  }
  A.cls_b = F(PB_CLS_B); A.cls_w = F(PB_CLS_W);
  A.conv1_b = F(PB_CONV1_B); A.conv1_w = F(PB_CONV1_W);
  return A;
}

static void run_branch(const float* mat, bool transA, const int* ei, const float* ew,
                       void* const* d_in, int base, Ws& W, float* x4out,
                       hipStream_t st) {
  auto F = [&](int idx) { return (const float*)d_in[base + idx]; };
  const int* src = ei;
  const int* dst = ei + EE;

  // --- degree / symmetric norm (self loops weight 1) ---
  k_fill_f32<<<blks(NN, 256), 256, 0, st>>>(W.dinv, 1.0f, NN);
  k_deg_edges<<<blks(EE, 256), 256, 0, st>>>(dst, ew, W.dinv, EE);
  k_dinv<<<blks(NN, 256), 256, 0, st>>>(W.dinv, NN);

  // --- GCN1: h = x @ W1 ; aggregate ; leaky(+b) -> x1 [N,256] ---
  launch_gemm(mat, F(PB_GCN1_W), W.B0, nullptr, NN, H1C, NN,
              transA ? 1 : NN, transA ? NN : 1, H1C, 1, st);
  k_gcn_selfinit<<<blks((long long)NN * H1C, 256), 256, 0, st>>>(W.B0, W.dinv, W.B1, 8, (long long)NN * H1C);
  k_gcn_agg<<<blks((long long)EE * H1C, 256), 256, 0, st>>>(W.B0, src, dst, ew, W.dinv, W.B1, 8, (long long)EE * H1C);
  k_gcn_final<<<blks((long long)NN * H1C, 256), 256, 0, st>>>(W.B1, F(PB_GCN1_B), W.B2, 8, (long long)NN * H1C);

  // --- GCN2: x1 @ W2 -> x2 [N,512] ---
  launch_gemm(W.B2, F(PB_GCN2_W), W.B0, nullptr, NN, H2C, H1C, H1C, 1, H2C, 1, st);
  k_gcn_selfinit<<<blks((long long)NN * H2C, 256), 256, 0, st>>>(W.B0, W.dinv, W.B1, 9, (long long)NN * H2C);
  k_gcn_agg<<<blks((long long)EE * H2C, 256), 256, 0, st>>>(W.B0, src, dst, ew, W.dinv, W.B1, 9, (long long)EE * H2C);
  k_gcn_final<<<blks((long long)NN * H2C, 256), 256, 0, st>>>(W.B1, F(PB_GCN2_B), W.B3, 9, (long long)NN * H2C);

  // --- DRSNet per node -> x3 [N,256] (incl. output leaky 0.01) ---
  k_drsnet<<<NN, 128, 0, st>>>(W.B3, W.B4, mk_drs(d_in, base));

  // --- GAT input s = x3 + x1 ---
  k_add<<<blks((long long)NN * H1C, 256), 256, 0, st>>>(W.B4, W.B2, W.B5, (long long)NN * H1C);

  // --- mean edge weight for self-loop edge_attr ---
  k_fill_f32<<<1, 32, 0, st>>>(W.ewsum, 0.0f, 1);
  k_reduce_sum<<<256, 256, 0, st>>>(ew, W.ewsum, EE);

  // --- xl / xr GEMMs (bias) ---
  launch_gemm(W.B5, F(PB_GAT_WL), W.B1, F(PB_GAT_BL), NN, HEADS * OUTC, H1C, H1C, 1, HEADS * OUTC, 1, st);
  launch_gemm(W.B5, F(PB_GAT_WR), W.B6, F(PB_GAT_BR), NN, HEADS * OUTC, H1C, H1C, 1, HEADS * OUTC, 1, st);

  // --- attention softmax + aggregation ---
  long long nt = (long long)(EE + NN) * HEADS;
  k_fill_u32<<<blks((long long)NN * HEADS, 256), 256, 0, st>>>(W.mmax, 0u, (long long)NN * HEADS);
  k_gat_logit<<<blks(nt, 256), 256, 0, st>>>(W.B1, W.B6, src, dst, ew, W.ewsum,
                                             F(PB_GAT_WE), F(PB_GAT_ATT), W.logit, W.mmax);
  k_fill_f32<<<blks((long long)NN * HEADS, 256), 256, 0, st>>>(W.ssum, 0.0f, (long long)NN * HEADS);
  k_fill_f32<<<blks((long long)NN * HEADS * OUTC, 256), 256, 0, st>>>(W.B0, 0.0f, (long long)NN * HEADS * OUTC);
  k_gat_accum<<<blks(nt, 256), 256, 0, st>>>(W.B1, src, dst, W.logit, W.mmax, W.ssum, W.B0);
  k_gat_final<<<blks((long long)NN * OUTC, 256), 256, 0, st>>>(W.B0, W.ssum, F(PB_GAT_B), x4out);
}

extern "C" void kernel_launch(void* const* d_in, const int* in_sizes, int n_in,
                              void* d_out, int out_size, void* d_ws, size_t ws_size,
                              hipStream_t stream) {
  (void)in_sizes; (void)n_in; (void)out_size; (void)ws_size;
  const float* mat    = (const float*)d_in[IN_MAT];
  const int* dis_ei   = (const int*)d_in[IN_DIS_EI];
  const int* drug_ei  = (const int*)d_in[IN_DRUG_EI];
  const float* dis_ew  = (const float*)d_in[IN_DIS_EW];
  const float* drug_ew = (const float*)d_in[IN_DRUG_EW];

  // workspace carve-out (256B aligned)
  size_t off = 0;
  auto take = [&](long long elems) {
    float* p = (float*)((char*)d_ws + off);
    off = (off + (size_t)elems * sizeof(float) + 255) & ~(size_t)255;
    return p;
  };
  Ws W;
  W.B0 = take((long long)NN * 512);  // gemm out / GAT acc
  W.B1 = take((long long)NN * 512);  // agg target / xl
  W.B2 = take((long long)NN * 256);  // x1
  W.B3 = take((long long)NN * 512);  // x2
  W.B4 = take((long long)NN * 256);  // x3
  W.B5 = take((long long)NN * 256);  // x3 + x1
  W.B6 = take((long long)NN * 512);  // xr
  W.dinv = take(NN);
  W.logit = take((long long)(EE + NN) * HEADS);
  W.mmax = (unsigned*)take((long long)NN * HEADS);
  W.ssum = take((long long)NN * HEADS);
  W.ewsum = take(64);
  W.x4dis = take((long long)NN * OUTC);
  W.x4drug = take((long long)NN * OUTC);

  // disease branch uses mat.T (strided access, no materialized transpose)
  run_branch(mat, /*transA=*/true, dis_ei, dis_ew, d_in, P_DIS, W, W.x4dis, stream);
  // drug branch uses mat
  run_branch(mat, /*transA=*/false, drug_ei, drug_ew, d_in, P_DRUG, W, W.x4drug, stream);

  // out = drug_x4 @ dis_x4.T  -> [8192, 8192]
  launch_gemm(W.x4drug, W.x4dis, (float*)d_out, nullptr, NN, NN, OUTC,
              OUTC, 1, /*ldb_k=*/1, /*ldb_n=*/OUTC, stream);
}